// RotationEstimator_8572754722877
// MI455X (gfx1250) — compile-verified
//
#include <hip/hip_runtime.h>
#include <stdint.h>

#define H_IMG 1024
#define W_IMG 1280
#define NPIX  (H_IMG * W_IMG)
#define N_ITER 5
#define TW 256
#define NBLK_C 1024

typedef float v8f __attribute__((ext_vector_type(8)));
typedef float v2f __attribute__((ext_vector_type(2)));

#if __has_builtin(__builtin_amdgcn_wmma_f32_16x16x4_f32)
#define HAS_WMMA_F32X4 1
#else
#define HAS_WMMA_F32X4 0
#endif

#if __has_builtin(__builtin_amdgcn_global_load_async_to_lds_b32)
#define HAS_ASYNC 1
typedef __attribute__((address_space(1))) int* gint_t;
typedef __attribute__((address_space(3))) int* lint_t;
#else
#define HAS_ASYNC 0
#endif

// ---------------------------------------------------------------------------
// 3x3 matmul D = A*B on the CDNA5 matrix pipe (V_WMMA_F32_16X16X4_F32).
// Inputs are lane-uniform; the 3x3 occupies the top-left of the 16x16x4 shape.
// A layout (ISA 7.12.2): lane<16 -> row M=lane, VGPR0=K0, VGPR1=K1;
//                        lane>=16 -> row M=lane-16, VGPR0=K2, VGPR1=K3.
// B layout (mirrored K split): lane<16 -> col N=lane, VGPR0=K0, VGPR1=K1;
//                              lane>=16 -> col N=lane-16, VGPR0=K2, VGPR1=K3.
// D layout: VGPR m holds row M=m (lanes 0-15), N = lane.
// Must be called with a full wave (32 threads), EXEC all ones.
// ---------------------------------------------------------------------------
__device__ __forceinline__ void mm3(const float* A, const float* B, float* D) {
#if HAS_WMMA_F32X4
  const int lane = (int)(threadIdx.x & 31u);
  const int m  = lane & 15;
  const int kh = lane >> 4;          // 0 -> K=0,1 ; 1 -> K=2,3
  const int k0 = kh * 2;
  const int k1 = k0 + 1;
  v2f av, bv;
  av.x = (m < 3)           ? A[m * 3 + k0] : 0.0f;   // k0 in {0,2} always < 3
  av.y = (m < 3 && k1 < 3) ? A[m * 3 + k1] : 0.0f;
  bv.x = (m < 3)           ? B[k0 * 3 + m] : 0.0f;
  bv.y = (m < 3 && k1 < 3) ? B[k1 * 3 + m] : 0.0f;
  v8f c = {0.f, 0.f, 0.f, 0.f, 0.f, 0.f, 0.f, 0.f};
  v8f d = __builtin_amdgcn_wmma_f32_16x16x4_f32(false, av, false, bv,
                                                (short)0, c, false, false);
  float dr0 = d[0], dr1 = d[1], dr2 = d[2];
  float row[3];
#pragma unroll
  for (int nn = 0; nn < 3; ++nn) {
    row[0] = __shfl(dr0, nn, 32);
    row[1] = __shfl(dr1, nn, 32);
    row[2] = __shfl(dr2, nn, 32);
    D[0 * 3 + nn] = row[0];
    D[1 * 3 + nn] = row[1];
    D[2 * 3 + nn] = row[2];
  }
#else
  float t[9];
#pragma unroll
  for (int mm = 0; mm < 3; ++mm)
#pragma unroll
    for (int nn = 0; nn < 3; ++nn)
      t[mm * 3 + nn] = A[mm * 3 + 0] * B[0 * 3 + nn] +
                       A[mm * 3 + 1] * B[1 * 3 + nn] +
                       A[mm * 3 + 2] * B[2 * 3 + nn];
#pragma unroll
  for (int i = 0; i < 9; ++i) D[i] = t[i];
#endif
}

// 3x3 inverse via adjugate / determinant.
__device__ __forceinline__ void inv3(const float* M, float* O) {
  float a = M[0], b = M[1], c = M[2];
  float d = M[3], e = M[4], f = M[5];
  float g = M[6], h = M[7], i = M[8];
  float C00 = e * i - f * h;
  float C01 = -(d * i - f * g);
  float C02 = d * h - e * g;
  float det = a * C00 + b * C01 + c * C02;
  float inv = 1.0f / det;
  O[0] = C00 * inv;
  O[1] = -(b * i - c * h) * inv;
  O[2] = (b * f - c * e) * inv;
  O[3] = C01 * inv;
  O[4] = (a * i - c * g) * inv;
  O[5] = -(a * f - c * d) * inv;
  O[6] = C02 * inv;
  O[7] = -(a * h - b * g) * inv;
  O[8] = (a * e - b * d) * inv;
}

// ---------------------------------------------------------------------------
// ws layout (floats): [0..8]=R, [9..17]=Hi, [32 .. 32+NBLK_C*9) = block partials
// ---------------------------------------------------------------------------

__global__ void rot_init(float* __restrict__ ws) {
  unsigned t = threadIdx.x;
  if (t < 9) ws[t] = (t == 0 || t == 4 || t == 8) ? 1.0f : 0.0f;
}

// One wave. Hi = inv( (K @ R) @ inv(K) ), matching reference evaluation order.
__global__ void rot_homog(const float* __restrict__ K9, float* __restrict__ ws) {
  float R[9], Kl[9];
#pragma unroll
  for (int i = 0; i < 9; ++i) { R[i] = ws[i]; Kl[i] = K9[i]; }
  float f0 = Kl[0], f1 = Kl[4], cu = Kl[2], cv = Kl[5];
  float Ki[9] = {1.0f / f0, 0.0f, -cu / f0,
                 0.0f, 1.0f / f1, -cv / f1,
                 0.0f, 0.0f, 1.0f};
  float T[9], Hm[9], Hi[9];
  mm3(Kl, R, T);    // K @ R        (WMMA)
  mm3(T, Ki, Hm);   // (K@R) @ K^-1 (WMMA)
  inv3(Hm, Hi);
  if (threadIdx.x < 9) ws[9 + threadIdx.x] = Hi[threadIdx.x];
}

// Bilinear warp with zero OOB taps + masked residual; writes d_out regions.
__global__ void __launch_bounds__(256) rot_warp(
    const float* __restrict__ ref, const float* __restrict__ tgt,
    const float* __restrict__ msk, const float* __restrict__ ws,
    float* __restrict__ warped, float* __restrict__ res) {
  int p = (int)(blockIdx.x * blockDim.x + threadIdx.x);
  if (p >= NPIX) return;
  const float* Hi = ws + 9;
  float h00 = Hi[0], h01 = Hi[1], h02 = Hi[2];
  float h10 = Hi[3], h11 = Hi[4], h12 = Hi[5];
  float h20 = Hi[6], h21 = Hi[7], h22 = Hi[8];
  int vy = p / W_IMG;
  int ux = p - vy * W_IMG;
  float u = (float)ux, v = (float)vy;
  float den = h20 * u + h21 * v + h22;
  float xs = (h00 * u + h01 * v + h02) / den;
  float ys = (h10 * u + h11 * v + h12) / den;
  float x0f = floorf(xs), y0f = floorf(ys);
  float wx = xs - x0f, wy = ys - y0f;
  int x0 = (int)x0f, y0 = (int)y0f;
  float val = 0.0f;
  {
    int xi = x0, yi = y0;
    float tp = (xi >= 0 && xi < W_IMG && yi >= 0 && yi < H_IMG)
                   ? ref[yi * W_IMG + xi] : 0.0f;
    val += tp * ((1.0f - wx) * (1.0f - wy));
  }
  {
    int xi = x0 + 1, yi = y0;
    float tp = (xi >= 0 && xi < W_IMG && yi >= 0 && yi < H_IMG)
                   ? ref[yi * W_IMG + xi] : 0.0f;
    val += tp * (wx * (1.0f - wy));
  }
  {
    int xi = x0, yi = y0 + 1;
    float tp = (xi >= 0 && xi < W_IMG && yi >= 0 && yi < H_IMG)
                   ? ref[yi * W_IMG + xi] : 0.0f;
    val += tp * ((1.0f - wx) * wy);
  }
  {
    int xi = x0 + 1, yi = y0 + 1;
    float tp = (xi >= 0 && xi < W_IMG && yi >= 0 && yi < H_IMG)
                   ? ref[yi * W_IMG + xi] : 0.0f;
    val += tp * (wx * wy);
  }
  warped[p] = val;
  res[p] = msk[p] * (val - tgt[p]);
}

// Sobel(warped)+Sobel(target) averaged, contracted with batch_proj_jac,
// accumulating JtJ (6) + Jtr (3). Tiles staged into LDS via CDNA5 async
// global->LDS copies; per-block partials written in a fixed (deterministic)
// reduction order — no float atomics.
__global__ void __launch_bounds__(256) rot_grad(
    const float* __restrict__ warped, const float* __restrict__ tgt,
    const float* __restrict__ res, const float* __restrict__ bpj,
    float* __restrict__ part) {
  __shared__ float sw[3][TW + 2];
  __shared__ float st[3][TW + 2];
  __shared__ float sred[8][9];
  float acc[9];
#pragma unroll
  for (int j = 0; j < 9; ++j) acc[j] = 0.0f;

  const int tilesPerRow = W_IMG / TW;            // 5
  const int ntiles = tilesPerRow * H_IMG;        // 5120
  for (int tile = (int)blockIdx.x; tile < ntiles; tile += NBLK_C) {
    int ty = tile / tilesPerRow;
    int tx0 = (tile - ty * tilesPerRow) * TW;
    if (ty < 2 || ty > H_IMG - 3) continue;      // uniform per block
    __syncthreads();                             // previous tile fully consumed
    for (int i = (int)threadIdx.x; i < 3 * (TW + 2); i += 256) {
      int r = i / (TW + 2);
      int cc = i - r * (TW + 2);
      int gy = ty - 1 + r;
      int gx = tx0 - 1 + cc;
      if (gx >= 0 && gx < W_IMG) {
        int idx = gy * W_IMG + gx;
#if HAS_ASYNC
        __builtin_amdgcn_global_load_async_to_lds_b32(
            (gint_t)(const void*)(warped + idx),
            (lint_t)(void*)&sw[r][cc], 0, 0);
        __builtin_amdgcn_global_load_async_to_lds_b32(
            (gint_t)(const void*)(tgt + idx),
            (lint_t)(void*)&st[r][cc], 0, 0);
#else
        sw[r][cc] = warped[idx];
        st[r][cc] = tgt[idx];
#endif
      } else {
        sw[r][cc] = 0.0f;
        st[r][cc] = 0.0f;
      }
    }
#if HAS_ASYNC
#if __has_builtin(__builtin_amdgcn_s_wait_asynccnt)
    __builtin_amdgcn_s_wait_asynccnt(0);
#else
    asm volatile("s_wait_asynccnt 0" ::: "memory");
#endif
#endif
    __syncthreads();

    int t = (int)threadIdx.x;
    int x = tx0 + t;
    if (x >= 2 && x <= W_IMG - 3) {
      // rows: sw[0]=y-1, sw[1]=y, sw[2]=y+1 ; cols: t=x-1, t+1=x, t+2=x+1
      float gxw = 0.125f * (sw[0][t + 2] - sw[0][t]) +
                  0.25f  * (sw[1][t + 2] - sw[1][t]) +
                  0.125f * (sw[2][t + 2] - sw[2][t]);
      float gyw = 0.125f * (sw[2][t] - sw[0][t]) +
                  0.25f  * (sw[2][t + 1] - sw[0][t + 1]) +
                  0.125f * (sw[2][t + 2] - sw[0][t + 2]);
      float gxt = 0.125f * (st[0][t + 2] - st[0][t]) +
                  0.25f  * (st[1][t + 2] - st[1][t]) +
                  0.125f * (st[2][t + 2] - st[2][t]);
      float gyt = 0.125f * (st[2][t] - st[0][t]) +
                  0.25f  * (st[2][t + 1] - st[0][t + 1]) +
                  0.125f * (st[2][t + 2] - st[0][t + 2]);
      float g0 = 0.5f * (gxw + gxt);
      float g1 = 0.5f * (gyw + gyt);
      int n = ty * W_IMG + x;
      const float* bp = bpj + (size_t)n * 6;
      float J0 = g0 * bp[0] + g1 * bp[3];
      float J1 = g0 * bp[1] + g1 * bp[4];
      float J2 = g0 * bp[2] + g1 * bp[5];
      float rr = res[n];
      acc[0] += J0 * J0; acc[1] += J0 * J1; acc[2] += J0 * J2;
      acc[3] += J1 * J1; acc[4] += J1 * J2; acc[5] += J2 * J2;
      acc[6] += J0 * rr; acc[7] += J1 * rr; acc[8] += J2 * rr;
    }
  }

  // Deterministic wave tree reduction, then fixed-order cross-wave sum.
#pragma unroll
  for (int j = 0; j < 9; ++j) {
    float a = acc[j];
#pragma unroll
    for (int o = 16; o >= 1; o >>= 1) a += __shfl_down(a, o, 32);
    acc[j] = a;
  }
  int wave = (int)(threadIdx.x >> 5);
  int lane = (int)(threadIdx.x & 31u);
  if (lane == 0) {
#pragma unroll
    for (int j = 0; j < 9; ++j) sred[wave][j] = acc[j];
  }
  __syncthreads();
  if (threadIdx.x < 9) {
    float s = 0.0f;
#pragma unroll
    for (int w2 = 0; w2 < 8; ++w2) s += sred[w2][threadIdx.x];
    part[blockIdx.x * 9 + threadIdx.x] = s;
  }
}

// One wave: deterministic final reduction, normal-equation solve
// (== pinv(J)@res for full-rank J), so3_exp, R update (2 WMMAs).
__global__ void rot_solve(const float* __restrict__ part, float* __restrict__ ws,
                          float* __restrict__ Rout) {
  int lane = (int)(threadIdx.x & 31u);
  float s[9];
#pragma unroll
  for (int j = 0; j < 9; ++j) {
    float a = 0.0f;
    for (int i = lane; i < NBLK_C; i += 32) a += part[i * 9 + j];
#pragma unroll
    for (int o = 16; o >= 1; o >>= 1) a += __shfl_down(a, o, 32);
    s[j] = __shfl(a, 0, 32);  // broadcast: lane-uniform
  }
  float A[9] = {s[0], s[1], s[2], s[1], s[3], s[4], s[2], s[4], s[5]};
  float Ai[9];
  inv3(A, Ai);
  float x0 = Ai[0] * s[6] + Ai[1] * s[7] + Ai[2] * s[8];
  float x1 = Ai[3] * s[6] + Ai[4] * s[7] + Ai[5] * s[8];
  float x2 = Ai[6] * s[6] + Ai[7] * s[7] + Ai[8] * s[8];
  float t2 = x0 * x0 + x1 * x1 + x2 * x2;
  float th = sqrtf(t2 + 1e-24f);
  float Af, Bf;
  if (th < 1e-4f) {  // uniform branch: EXEC unchanged
    Af = 1.0f - t2 / 6.0f;
    Bf = 0.5f - t2 / 24.0f;
  } else {
    Af = sinf(th) / th;
    Bf = (1.0f - cosf(th)) / (t2 + 1e-24f);
  }
  float Km[9] = {0.0f, -x2, x1, x2, 0.0f, -x0, -x1, x0, 0.0f};
  float K2[9];
  mm3(Km, Km, K2);                 // Km @ Km (WMMA)
  float E[9];
#pragma unroll
  for (int i = 0; i < 9; ++i)
    E[i] = ((i % 4 == 0) ? 1.0f : 0.0f) + Af * Km[i] + Bf * K2[i];
  float Rc[9];
#pragma unroll
  for (int i = 0; i < 9; ++i) Rc[i] = ws[i];
  float Rn[9];
  mm3(Rc, E, Rn);                  // R @ exp(x^) (WMMA)
  if (lane < 9) {
    ws[lane] = Rn[lane];
    Rout[lane] = Rn[lane];
  }
}

extern "C" void kernel_launch(void* const* d_in, const int* in_sizes, int n_in,
                              void* d_out, int out_size, void* d_ws, size_t ws_size,
                              hipStream_t stream) {
  (void)in_sizes; (void)n_in; (void)out_size; (void)ws_size;
  const float* ref = (const float*)d_in[0];
  const float* tgt = (const float*)d_in[1];
  const float* msk = (const float*)d_in[2];
  const float* K9  = (const float*)d_in[3];
  const float* bpj = (const float*)d_in[4];
  float* out = (float*)d_out;
  float* Rout = out;                 // 9
  float* res  = out + 9;             // NPIX
  float* wrp  = out + 9 + NPIX;      // NPIX
  float* ws   = (float*)d_ws;
  float* part = ws + 32;             // NBLK_C * 9 floats (~36 KB)

  rot_init<<<1, 32, 0, stream>>>(ws);
  for (int it = 0; it < N_ITER; ++it) {
    rot_homog<<<1, 32, 0, stream>>>(K9, ws);
    rot_warp<<<NPIX / 256, 256, 0, stream>>>(ref, tgt, msk, ws, wrp, res);
    rot_grad<<<NBLK_C, 256, 0, stream>>>(wrp, tgt, res, bpj, part);
    rot_solve<<<1, 32, 0, stream>>>(part, ws, Rout);
  }
}